// JpegLayer_50611894616615
// MI455X (gfx1250) — compile-verified
//
#include <hip/hip_runtime.h>
#include <math.h>

typedef float v2f __attribute__((ext_vector_type(2)));
typedef float v8f __attribute__((ext_vector_type(8)));

#define LDS_STRIDE 18   // 16 + 2 pad: row j and row j+8 hit disjoint bank runs

// cos(t*pi/16), t = 0..8 (double-precision derived)
__constant__ float COS16[9] = {
    1.0f, 0.9807852804032304f, 0.9238795325112867f, 0.8314696123025452f,
    0.7071067811865476f, 0.5555702330196022f, 0.3826834323650898f,
    0.1950903220161283f, 0.0f};

// D[k][n] = 2*cos(pi*(2n+1)*k/16)
__device__ __forceinline__ float Dmat(int k, int n) {
  int t = ((2 * n + 1) * k) & 31;   // periodic in 32
  float s = 1.0f;
  if (t > 16) t = 32 - t;           // cos symmetric about pi
  if (t > 8) { s = -1.0f; t = 16 - t; }
  return 2.0f * s * COS16[t];
}

// row scale for D^{-1} = D^T * diag(1/32, 1/16, ..., 1/16)
__device__ __forceinline__ float sK(int k) {
  return (k == 0) ? (1.0f / 32.0f) : (1.0f / 16.0f);
}

__global__ __launch_bounds__(256) void jpeg_wmma_kernel(
    const float* __restrict__ x, const float* __restrict__ quant,
    float* __restrict__ out, int totalWaves) {
  constexpr int H = 1024, W = 1024;
  constexpr size_t plane = (size_t)H * W;   // 1M floats (4 MiB)
  constexpr int tilesX = W >> 4;            // 64
  constexpr int tpb = tilesX * (H >> 4);    // 4096 tiles per batch image

  __shared__ float lds[8][16 * LDS_STRIDE];

  const int lane = threadIdx.x & 31;
  const int wave = threadIdx.x >> 5;
  const int hi   = lane >> 4;     // 0 = lanes 0-15, 1 = lanes 16-31
  const int ln   = lane & 15;
  float* tile = lds[wave];

  int w = blockIdx.x * 8 + wave;
  if (w >= totalWaves) return;
  int b   = w / tpb;              // constant divisor -> shift
  int rem = w - b * tpb;
  int ty  = rem / tilesX;         // constant divisor -> shift
  int tx  = rem - ty * tilesX;

  const size_t base = (size_t)b * 3 * plane + (size_t)(ty << 4) * W + (size_t)(tx << 4);

  // ---- per-lane constants ------------------------------------------------
  // Quant for C-layout slot j: element (m=j+8*hi, n=ln) -> QUANT[m&7][n&7] = QUANT[j][ln&7]
  float qv[8], rv[8];
#pragma unroll
  for (int j = 0; j < 8; ++j) {
    qv[j] = quant[j * 8 + (ln & 7)];
    rv[j] = __builtin_amdgcn_rcpf(qv[j]);   // v_rcp_f32, 1-ULP on small ints
  }

  // dF: D2 as A-operand at (m=ln,k) == D2^T as B-operand at (k,n=ln)  (same value)
  // dI: Di2 as A-operand       == Di2^T as B-operand                  (same value)
  float dF[8], dI[8];
#pragma unroll
  for (int v = 0; v < 8; ++v) {
    int k = 4 * (v >> 1) + (v & 1) + 2 * hi;       // K index held by this vgpr slot
    bool same = ((ln >> 3) == (k >> 3));           // block-diagonal mask
    dF[v] = same ? Dmat(ln & 7, k & 7) : 0.0f;     // D2[ln][k]
    dI[v] = same ? (Dmat(k & 7, ln & 7) * sK(k & 7)) : 0.0f;  // Di2[ln][k]
  }

  // ---- load RGB tile (NT), convert to YCbCr ------------------------------
  // B-operand layout: slot v holds element (row = 4*(v/2)+(v&1)+2*hi, col = ln).
  // Fold ln and the lane-half row offset (2*hi) into the base pointer so the
  // unrolled per-slot offsets (row*W, +plane) are constant instruction imms.
  const float* pxR = x + (base + (size_t)ln + (size_t)(2 * hi) * W);
  const float* pxB = pxR + 2 * plane;   // 2*plane*4 bytes just exceeds imm24
  float X[3][8];
#pragma unroll
  for (int v = 0; v < 8; ++v) {
    int row = 4 * (v >> 1) + (v & 1);             // compile-time constant
    float r  = __builtin_nontemporal_load(pxR + row * W);
    float g  = __builtin_nontemporal_load(pxR + plane + row * W);
    float bl = __builtin_nontemporal_load(pxB + row * W);
    X[0][v] =  0.299f  * r + 0.587f  * g + 0.114f * bl - 128.0f;
    X[1][v] = -0.1687f * r - 0.3313f * g + 0.5f   * bl;
    X[2][v] =  0.5f    * r - 0.4187f * g - 0.0813f * bl;
  }

  float Z[3][8];
#pragma unroll
  for (int c = 0; c < 3; ++c) {
    // stage 1: T = D2 @ X      (A = dF const, B = X regs)
    v8f acc1 = {0.f,0.f,0.f,0.f,0.f,0.f,0.f,0.f};
#pragma unroll
    for (int ch = 0; ch < 4; ++ch) {
      v2f a = {dF[2 * ch], dF[2 * ch + 1]};
      v2f bb = {X[c][2 * ch], X[c][2 * ch + 1]};
      acc1 = __builtin_amdgcn_wmma_f32_16x16x4_f32(false, a, false, bb,
                                                   (short)0, acc1, false, false);
    }
    // C layout -> LDS (m = j+8*hi, n = ln)
#pragma unroll
    for (int j = 0; j < 8; ++j) tile[(j + 8 * hi) * LDS_STRIDE + ln] = acc1[j];
    asm volatile("s_wait_dscnt 0" ::: "memory");

    // stage 2: Y = T @ D2^T    (A = T from LDS in A-layout, B = dF)
    v8f acc2 = {0.f,0.f,0.f,0.f,0.f,0.f,0.f,0.f};
#pragma unroll
    for (int ch = 0; ch < 4; ++ch) {
      int k0 = 4 * ch + 2 * hi;
      v2f a = {tile[ln * LDS_STRIDE + k0], tile[ln * LDS_STRIDE + k0 + 1]};
      v2f bb = {dF[2 * ch], dF[2 * ch + 1]};
      acc2 = __builtin_amdgcn_wmma_f32_16x16x4_f32(false, a, false, bb,
                                                   (short)0, acc2, false, false);
    }
    asm volatile("s_wait_dscnt 0" ::: "memory");  // WAR: loads done before rewrite

    // quantize (trunc toward zero) + dequantize, scatter to LDS
#pragma unroll
    for (int j = 0; j < 8; ++j) {
      float yq = truncf(acc2[j] * rv[j]) * qv[j];
      tile[(j + 8 * hi) * LDS_STRIDE + ln] = yq;
    }
    asm volatile("s_wait_dscnt 0" ::: "memory");

    // stage 3: U = Di2 @ Yq    (A = dI, B = Yq from LDS in B-layout)
    v8f acc3 = {0.f,0.f,0.f,0.f,0.f,0.f,0.f,0.f};
#pragma unroll
    for (int ch = 0; ch < 4; ++ch) {
      int k0 = 4 * ch + 2 * hi;
      v2f a = {dI[2 * ch], dI[2 * ch + 1]};
      v2f bb = {tile[k0 * LDS_STRIDE + ln], tile[(k0 + 1) * LDS_STRIDE + ln]};
      acc3 = __builtin_amdgcn_wmma_f32_16x16x4_f32(false, a, false, bb,
                                                   (short)0, acc3, false, false);
    }
    asm volatile("s_wait_dscnt 0" ::: "memory");
#pragma unroll
    for (int j = 0; j < 8; ++j) tile[(j + 8 * hi) * LDS_STRIDE + ln] = acc3[j];
    asm volatile("s_wait_dscnt 0" ::: "memory");

    // stage 4: Z = U @ Di2^T   (A = U from LDS in A-layout, B = dI)
    v8f acc4 = {0.f,0.f,0.f,0.f,0.f,0.f,0.f,0.f};
#pragma unroll
    for (int ch = 0; ch < 4; ++ch) {
      int k0 = 4 * ch + 2 * hi;
      v2f a = {tile[ln * LDS_STRIDE + k0], tile[ln * LDS_STRIDE + k0 + 1]};
      v2f bb = {dI[2 * ch], dI[2 * ch + 1]};
      acc4 = __builtin_amdgcn_wmma_f32_16x16x4_f32(false, a, false, bb,
                                                   (short)0, acc4, false, false);
    }
#pragma unroll
    for (int j = 0; j < 8; ++j) Z[c][j] = acc4[j];
    asm volatile("s_wait_dscnt 0" ::: "memory");  // protect LDS reuse next channel
  }

  // ---- YCbCr -> RGB (+128 on Y; Cb/Cr offsets cancel), med3 clamp, NT store
  // C-layout row = j + 8*hi: fold ln and 8*hi into the output base pointers.
  float* poR = out + (base + (size_t)ln + (size_t)(8 * hi) * W);
  float* poB = poR + 2 * plane;
#pragma unroll
  for (int j = 0; j < 8; ++j) {
    float yy = Z[0][j] + 128.0f;
    float cb = Z[1][j];
    float cr = Z[2][j];
    float r  = yy + 1.402f * cr;
    float g  = yy + 0.34414f * cb - 0.71414f * cr;  // matches reference matrix
    float bl = yy + 1.772f * cb;
    r  = __builtin_amdgcn_fmed3f(r,  0.0f, 255.0f);
    g  = __builtin_amdgcn_fmed3f(g,  0.0f, 255.0f);
    bl = __builtin_amdgcn_fmed3f(bl, 0.0f, 255.0f);
    __builtin_nontemporal_store(r,  poR + j * W);
    __builtin_nontemporal_store(g,  poR + plane + j * W);
    __builtin_nontemporal_store(bl, poB + j * W);
  }
}

extern "C" void kernel_launch(void* const* d_in, const int* in_sizes, int n_in,
                              void* d_out, int out_size, void* d_ws, size_t ws_size,
                              hipStream_t stream) {
  const float* x = (const float*)d_in[0];
  const float* q = (const float*)d_in[1];
  float* out = (float*)d_out;

  const int H = 1024, W = 1024;
  int B = in_sizes[0] / (3 * H * W);          // 16
  int totalWaves = B * (H >> 4) * (W >> 4);   // 65536 tiles, 1 wave each
  int blocks = (totalWaves + 7) / 8;          // 8 waves (256 thr) per block
  jpeg_wmma_kernel<<<blocks, 256, 0, stream>>>(x, q, out, totalWaves);
}